// LeadSpecificPatchPool_new_11914239279609
// MI455X (gfx1250) — compile-verified
//
#include <hip/hip_runtime.h>
#include <hip/hip_bf16.h>

typedef float v2f __attribute__((ext_vector_type(2)));
typedef float v8f __attribute__((ext_vector_type(8)));

// Problem constants
#define BB 64
#define NN 768
#define FF 256
#define PP 64
#define TT 16
#define KK 3
#define LL 12   // N / P

// ---------------------------------------------------------------------------
// Kernel A: pooled_x[b, t*L+l, f] = sum_{p,k} x[b, p*L+l, f+k-1]*w[t,p,k] + b[t]
// One workgroup per (b,l). 256 threads = 8 waves; each wave computes 16x16
// output tiles (t x f) with V_WMMA_F32_16X16X4_F32, K-dim = p (64).
// ---------------------------------------------------------------------------
__global__ __launch_bounds__(256) void pooled_x_kernel(
    const float* __restrict__ x, const float* __restrict__ w,
    const float* __restrict__ bias, float* __restrict__ out)
{
    __shared__ float Xl[PP * FF];   // 64 KB: X_l[p][f]

    const int bl = blockIdx.x;      // 0..767
    const int b  = bl / LL;
    const int l  = bl % LL;
    const int tid = threadIdx.x;

    // Stage X_l into LDS: row p comes from x[b, p*12 + l, :]
    const float* xb = x + (size_t)b * NN * FF;
    for (int i = tid; i < PP * FF; i += 256) {
        const int p = i >> 8;
        const int f = i & 255;
        Xl[i] = xb[(size_t)(p * LL + l) * FF + f];
    }
    __syncthreads();

    const int wave = tid >> 5;      // 0..7
    const int ln   = tid & 31;
    const int half = ln >> 4;       // 0/1
    const int col  = ln & 15;       // 0..15

    // Hoist A-fragments (weights) into registers: a.x = w[t=col, p=p0],
    // a.y = w[t=col, p0+1]; VGPR0 holds K offsets 0/2 split by half-wave.
    v2f afrag[KK][16];
#pragma unroll
    for (int k = 0; k < KK; ++k) {
#pragma unroll
        for (int p4 = 0; p4 < 16; ++p4) {
            const int p0 = p4 * 4 + 2 * half;
            afrag[k][p4].x = w[(col * PP + p0) * KK + k];
            afrag[k][p4].y = w[(col * PP + p0 + 1) * KK + k];
        }
    }

    for (int tile = wave; tile < FF / 16; tile += 8) {
        const int fbase = tile * 16;
        v8f acc = {};
#pragma unroll
        for (int k = 0; k < KK; ++k) {
            const int f  = fbase + col + k - 1;
            const int fc = f < 0 ? 0 : (f > FF - 1 ? FF - 1 : f);
            const float msk = (f == fc) ? 1.0f : 0.0f;  // zero-pad boundary
#pragma unroll
            for (int p4 = 0; p4 < 16; ++p4) {
                const int p0 = p4 * 4 + 2 * half;
                v2f bf;
                bf.x = Xl[p0 * FF + fc] * msk;
                bf.y = Xl[(p0 + 1) * FF + fc] * msk;
                acc = __builtin_amdgcn_wmma_f32_16x16x4_f32(
                    false, afrag[k][p4], false, bf, (short)0, acc, false, false);
            }
        }
        // D layout: acc[i] -> t = i + 8*half, f = fbase + col
#pragma unroll
        for (int i = 0; i < 8; ++i) {
            const int t = i + 8 * half;
            out[((size_t)b * (TT * LL) + t * LL + l) * FF + fbase + col] =
                acc[i] + bias[t];
        }
    }
}

// ---------------------------------------------------------------------------
// Kernel B: pooled_adj[b, l1*16+t, l2*16+u] = sum_{p,q} s[t,p]*A[p,q]*s[u,q]
// with s[t,p] = sum_k w[t,p,k]*r[k].  One workgroup per (b,l1); 384 threads
// = 12 waves, wave i handles l2 = i.  Stage 1: M1 = S x A (4 q-tiles),
// stage 2: Z = M1 x S^T via per-wave LDS transpose of M1.
// ---------------------------------------------------------------------------
#define M1STRIDE 65   // pad 16x64 tile rows to kill LDS bank conflicts

__global__ __launch_bounds__(384) void pooled_adj_kernel(
    const float* __restrict__ adj, const float* __restrict__ w,
    const float* __restrict__ re, float* __restrict__ out)
{
    __shared__ float Slds[TT * PP];            // 4 KB   s[t][p]
    __shared__ float M1[LL][TT * M1STRIDE];    // 48.75 KB, per-wave tiles

    const int bl = blockIdx.x;                 // 0..767
    const int b  = bl / LL;
    const int l1 = bl % LL;
    const int tid = threadIdx.x;               // 0..383

    // s[t,p] = sum_k w[t,p,k] * r[k]
    const float r0 = re[0], r1 = re[1], r2 = re[2];
    for (int i = tid; i < TT * PP; i += 384)
        Slds[i] = w[i * KK] * r0 + w[i * KK + 1] * r1 + w[i * KK + 2] * r2;
    __syncthreads();

    const int wave = tid >> 5;                 // 0..11 == l2
    const int ln   = tid & 31;
    const int half = ln >> 4;
    const int col  = ln & 15;
    const int l2   = wave;

    // S fragments serve as both stage-1 A operand and stage-2 B operand:
    // sf[p4].x = s[row=col, p = p4*4 + 2*half], .y = +1
    v2f sf[16];
#pragma unroll
    for (int p4 = 0; p4 < 16; ++p4) {
        const int p0 = p4 * 4 + 2 * half;
        sf[p4].x = Slds[col * PP + p0];
        sf[p4].y = Slds[col * PP + p0 + 1];
    }

    // ---- stage 1: M1[t,q] = sum_p s[t,p] * adj_blk[p,q], q in 4 tiles of 16
    const float* ab = adj + ((size_t)b * NN + l1 * PP) * NN + (size_t)l2 * PP;
    v8f m0 = {}, m1 = {}, m2 = {}, m3 = {};
#pragma unroll
    for (int p4 = 0; p4 < 16; ++p4) {
        const int p0 = p4 * 4 + 2 * half;
        const float* r0p = ab + (size_t)p0 * NN;
        const float* r1p = ab + (size_t)(p0 + 1) * NN;
        v2f bf;
        bf.x = r0p[col];        bf.y = r1p[col];
        m0 = __builtin_amdgcn_wmma_f32_16x16x4_f32(false, sf[p4], false, bf, (short)0, m0, false, false);
        bf.x = r0p[16 + col];   bf.y = r1p[16 + col];
        m1 = __builtin_amdgcn_wmma_f32_16x16x4_f32(false, sf[p4], false, bf, (short)0, m1, false, false);
        bf.x = r0p[32 + col];   bf.y = r1p[32 + col];
        m2 = __builtin_amdgcn_wmma_f32_16x16x4_f32(false, sf[p4], false, bf, (short)0, m2, false, false);
        bf.x = r0p[48 + col];   bf.y = r1p[48 + col];
        m3 = __builtin_amdgcn_wmma_f32_16x16x4_f32(false, sf[p4], false, bf, (short)0, m3, false, false);
    }

    // Spill M1 to per-wave LDS tile in [t][q] order (C-layout -> addressable)
    float* mw = &M1[wave][0];
#pragma unroll
    for (int i = 0; i < 8; ++i) {
        const int t = i + 8 * half;
        mw[t * M1STRIDE + col]      = m0[i];
        mw[t * M1STRIDE + 16 + col] = m1[i];
        mw[t * M1STRIDE + 32 + col] = m2[i];
        mw[t * M1STRIDE + 48 + col] = m3[i];
    }
    // Same-wave DS ordering; make it explicit and keep the compiler honest.
    asm volatile("s_wait_dscnt 0" ::: "memory");

    // ---- stage 2: Z[t,u] = sum_q M1[t,q] * s[u,q]   (A = M1, B = S^T)
    v8f z = {};
#pragma unroll
    for (int q4 = 0; q4 < 16; ++q4) {
        const int q0 = q4 * 4 + 2 * half;
        v2f af;
        af.x = mw[col * M1STRIDE + q0];
        af.y = mw[col * M1STRIDE + q0 + 1];
        z = __builtin_amdgcn_wmma_f32_16x16x4_f32(false, af, false, sf[q4], (short)0, z, false, false);
    }

    // z[i]: t = i + 8*half, u = col -> out[b, l1*16+t, l2*16+u]; 16-float
    // contiguous segments per row => coalesced.
    float* ob = out + ((size_t)b * (LL * TT) + l1 * TT) * (LL * TT) + l2 * TT;
#pragma unroll
    for (int i = 0; i < 8; ++i) {
        const int t = i + 8 * half;
        ob[(size_t)t * (LL * TT) + col] = z[i];
    }
}

extern "C" void kernel_launch(void* const* d_in, const int* in_sizes, int n_in,
                              void* d_out, int out_size, void* d_ws, size_t ws_size,
                              hipStream_t stream) {
    const float* x    = (const float*)d_in[0];  // [64,768,256]
    const float* adj  = (const float*)d_in[1];  // [64,768,768]
    const float* w    = (const float*)d_in[2];  // [16,64,1,3]
    const float* bias = (const float*)d_in[3];  // [16]
    const float* re   = (const float*)d_in[4];  // [3,1]

    float* out_x   = (float*)d_out;                                  // 64*192*256
    float* out_adj = out_x + (size_t)BB * (TT * LL) * FF;            // 64*192*192

    pooled_x_kernel<<<dim3(BB * LL), dim3(256), 0, stream>>>(x, w, bias, out_x);
    pooled_adj_kernel<<<dim3(BB * LL), dim3(384), 0, stream>>>(adj, w, re, out_adj);
}